// quantized_CSI_feedback_36627481101242
// MI455X (gfx1250) — compile-verified
//
#include <hip/hip_runtime.h>

typedef float v2f __attribute__((ext_vector_type(2)));
typedef float v8f __attribute__((ext_vector_type(8)));

#define NRB 256
#define NC  2048
#define NVAR 0.1f          // 1 / 10^(10dB/10)
#define A_INFO 0.83f
#define B_INFO 0.73f

// workspace layout (float offsets)
#define OFF_HRE 0           // [rb][n][t]  256*2*4
#define OFF_HIM 2048
#define OFF_WR  4096        // [s][c][t]   2*2048*4
#define OFF_WI  20480
#define OFF_WNG 36864       // -Wi, same layout
#define OFF_PACK 53248      // 256 x u64 (8B aligned: 53248*4 % 8 == 0)

// ---------------- complex helpers ----------------
__device__ __forceinline__ float2 cmul(float2 a, float2 b) {
    return make_float2(a.x*b.x - a.y*b.y, a.x*b.y + a.y*b.x);
}
__device__ __forceinline__ float2 cmulc(float2 a, float2 b) {   // a * conj(b)
    return make_float2(a.x*b.x + a.y*b.y, a.y*b.x - a.x*b.y);
}
__device__ __forceinline__ float2 cadd(float2 a, float2 b) {
    return make_float2(a.x + b.x, a.y + b.y);
}
__device__ __forceinline__ float2 conjf2(float2 a) { return make_float2(a.x, -a.y); }
__device__ __forceinline__ float  abs2(float2 a)   { return a.x*a.x + a.y*a.y; }

// ---------------- kernel 1: RB reduction + precoder normalization + atomic init ----------------
__global__ void prep_kernel(const float* __restrict__ h_re, const float* __restrict__ h_im,
                            const float* __restrict__ cb_re, const float* __restrict__ cb_im,
                            float* __restrict__ ws) {
    int tid = blockIdx.x * blockDim.x + threadIdx.x;
    if (tid < 2048) {
        // H[rb][n][t] = mean over 14 symbols x 12 subcarriers
        int rb = tid >> 3, n = (tid >> 2) & 1, t = tid & 3;
        const float* pr = h_re + (size_t)((n*4 + t) * 14) * 3072 + rb * 12;
        const float* pi = h_im + (size_t)((n*4 + t) * 14) * 3072 + rb * 12;
        float sr = 0.f, si = 0.f;
        for (int sym = 0; sym < 14; ++sym) {
            const float4* vr = (const float4*)(pr + sym * 3072);
            const float4* vi = (const float4*)(pi + sym * 3072);
#pragma unroll
            for (int j = 0; j < 3; ++j) {
                float4 a = vr[j], b = vi[j];
                sr += (a.x + a.y) + (a.z + a.w);
                si += (b.x + b.y) + (b.z + b.w);
            }
        }
        const float scale = 1.0f / 168.0f;
        ws[OFF_HRE + (rb*2 + n)*4 + t] = sr * scale;
        ws[OFF_HIM + (rb*2 + n)*4 + t] = si * scale;
    } else if (tid < 4096) {
        // unit-Frobenius-norm precoders, stored WMMA-A-friendly: [s][c][t]
        int c = tid - 2048;
        float s2 = 0.f;
#pragma unroll
        for (int k = 0; k < 8; ++k) {
            float r = cb_re[c*8 + k], i = cb_im[c*8 + k];
            s2 += r*r + i*i;
        }
        float inv = rsqrtf(s2);
#pragma unroll
        for (int t = 0; t < 4; ++t)
#pragma unroll
            for (int s = 0; s < 2; ++s) {
                float r = cb_re[c*8 + t*2 + s] * inv;
                float i = cb_im[c*8 + t*2 + s] * inv;
                int idx = (s*NC + c)*4 + t;
                ws[OFF_WR  + idx] =  r;
                ws[OFF_WI  + idx] =  i;
                ws[OFF_WNG + idx] = -i;
            }
    } else if (tid < 4096 + NRB) {
        ((unsigned long long*)(ws + OFF_PACK))[tid - 4096] = 0ull;
    }
}

// ---------------- kernel 2: WMMA h_eff + MMSE SINR + packed atomic argmax ----------------
// One wave per tile: 16 codebooks x 8 RBs. Column n' = rb_local*2 + n, K = t.
__global__ void __launch_bounds__(32) pmi_wmma_kernel(const float* __restrict__ ws,
                                                      unsigned long long* __restrict__ packed) {
    __shared__ float lds_h[16 * 16 * 4];   // [m][n'][s][re/im]
    __shared__ float lds_rate[128];        // [m][rb_local]

    const int lane = threadIdx.x;
    const int cb_tile = blockIdx.x & 127;
    const int rb_tile = blockIdx.x >> 7;
    const int c0  = cb_tile * 16;
    const int rb0 = rb_tile * 8;
    const int half = lane >> 4;      // K half: lanes 16-31 hold K=2,3
    const int lm   = lane & 15;
    const int kk   = half * 2;

    // A operands (16x4): row m=lm, K=kk..kk+1
    v2f awr[2], awi[2], awn[2];
#pragma unroll
    for (int s = 0; s < 2; ++s) {
        int idx = (s*NC + (c0 + lm))*4 + kk;
        awr[s] = *(const v2f*)(ws + OFF_WR  + idx);
        awi[s] = *(const v2f*)(ws + OFF_WI  + idx);
        awn[s] = *(const v2f*)(ws + OFF_WNG + idx);
    }
    // B operand (4x16): column n'=lm -> (rb, n), rows K=kk..kk+1
    int hidx = ((rb0 + (lm >> 1))*2 + (lm & 1))*4 + kk;
    v2f bhr = *(const v2f*)(ws + OFF_HRE + hidx);
    v2f bhi = *(const v2f*)(ws + OFF_HIM + hidx);

    v8f zero = {};
    v8f acc_re[2], acc_im[2];
#pragma unroll
    for (int s = 0; s < 2; ++s) {
        // (Wr + iWi)(Hr + iHi): re = Wr*Hr + (-Wi)*Hi ; im = Wi*Hr + Wr*Hi
        v8f re = __builtin_amdgcn_wmma_f32_16x16x4_f32(false, awr[s], false, bhr, (short)0, zero, false, false);
        re     = __builtin_amdgcn_wmma_f32_16x16x4_f32(false, awn[s], false, bhi, (short)0, re,   false, false);
        v8f im = __builtin_amdgcn_wmma_f32_16x16x4_f32(false, awi[s], false, bhr, (short)0, zero, false, false);
        im     = __builtin_amdgcn_wmma_f32_16x16x4_f32(false, awr[s], false, bhi, (short)0, im,   false, false);
        acc_re[s] = re;
        acc_im[s] = im;
    }

    // spill D to LDS: lane holds column lm, rows m = v + 8*half
#pragma unroll
    for (int v = 0; v < 8; ++v) {
        int m = v + 8 * half;
#pragma unroll
        for (int s = 0; s < 2; ++s) {
            float2 val = make_float2(acc_re[s][v], acc_im[s][v]);
            *(float2*)&lds_h[(m*16 + lm)*4 + s*2] = val;
        }
    }
    __syncthreads();

    // 128 (codebook, RB) pairs -> 4 per lane: 2x2 MMSE -> SINR -> rate
#pragma unroll
    for (int i = 0; i < 4; ++i) {
        int p = i * 32 + lane;
        int m = p >> 3, rbl = p & 7;
        float2 h00 = *(const float2*)&lds_h[(m*16 + (rbl*2 + 0))*4 + 0];
        float2 h01 = *(const float2*)&lds_h[(m*16 + (rbl*2 + 0))*4 + 2];
        float2 h10 = *(const float2*)&lds_h[(m*16 + (rbl*2 + 1))*4 + 0];
        float2 h11 = *(const float2*)&lds_h[(m*16 + (rbl*2 + 1))*4 + 2];

        // A = Heff Heff^H + nvar I  (2x2 Hermitian), analytic inverse
        float  a00 = abs2(h00) + abs2(h01) + NVAR;
        float  a11 = abs2(h10) + abs2(h11) + NVAR;
        float2 a01 = cadd(cmulc(h00, h10), cmulc(h01, h11));
        float  det = a00 * a11 - abs2(a01);
        float  id  = 1.0f / det;
        float  i00 = a11 * id, i11 = a00 * id;
        float2 i01 = make_float2(-a01.x * id, -a01.y * id);
        float2 i10 = make_float2(-a01.x * id,  a01.y * id);

        // G = Heff^H A^-1
        float2 ch00 = conjf2(h00), ch10 = conjf2(h10), ch01 = conjf2(h01), ch11 = conjf2(h11);
        float2 G00 = cadd(make_float2(ch00.x*i00, ch00.y*i00), cmul(ch10, i10));
        float2 G01 = cadd(cmul(ch00, i01), make_float2(ch10.x*i11, ch10.y*i11));
        float2 G10 = cadd(make_float2(ch01.x*i00, ch01.y*i00), cmul(ch11, i10));
        float2 G11 = cadd(cmul(ch01, i01), make_float2(ch11.x*i11, ch11.y*i11));

        // T = G Heff
        float2 T00 = cadd(cmul(G00, h00), cmul(G01, h10));
        float2 T01 = cadd(cmul(G00, h01), cmul(G01, h11));
        float2 T10 = cadd(cmul(G10, h00), cmul(G11, h10));
        float2 T11 = cadd(cmul(G10, h01), cmul(G11, h11));

        float n0 = NVAR * (abs2(G00) + abs2(G01));
        float n1 = NVAR * (abs2(G10) + abs2(G11));
        float sinr0 = abs2(T00) / (abs2(T01) + n0);
        float sinr1 = abs2(T11) / (abs2(T10) + n1);
        // EESM over singleton axis is identity -> rate directly
        float rate = A_INFO * (__log2f(1.0f + B_INFO * sinr0) +
                               __log2f(1.0f + B_INFO * sinr1));
        lds_rate[p] = rate;
    }
    __syncthreads();

    // per-RB local argmax over 16 codebooks, commit via packed u64 atomic max.
    // rate >= 0 so float bits are monotone as uint; low32 = ~c breaks ties toward smallest c (first max).
    if (lane < 8) {
        float best = -1.0f;
        int bm = 0;
        for (int m = 0; m < 16; ++m) {
            float v = lds_rate[m*8 + lane];
            if (v > best) { best = v; bm = m; }
        }
        unsigned long long pk = ((unsigned long long)__float_as_uint(best) << 32)
                              | (unsigned int)(~(unsigned)(c0 + bm));
        atomicMax(&packed[rb0 + lane], pk);
    }
}

// ---------------- kernel 3: unpack PMI/rate, gather selected precoders ----------------
__global__ void finalize_kernel(const float* __restrict__ ws,
                                int* __restrict__ pmi_out,
                                float* __restrict__ rate_out,
                                float* __restrict__ w_out) {
    int rb = blockIdx.x * blockDim.x + threadIdx.x;
    if (rb >= NRB) return;
    unsigned long long pk = ((const unsigned long long*)(ws + OFF_PACK))[rb];
    unsigned c = ~(unsigned)(pk & 0xFFFFFFFFull);
    pmi_out[rb]  = (int)c;
    rate_out[rb] = __uint_as_float((unsigned)(pk >> 32));
#pragma unroll
    for (int t = 0; t < 4; ++t)
#pragma unroll
        for (int s = 0; s < 2; ++s) {
            int idx = (s*NC + (int)c)*4 + t;
            w_out[rb*16 + (t*2 + s)*2 + 0] = ws[OFF_WR + idx];
            w_out[rb*16 + (t*2 + s)*2 + 1] = ws[OFF_WI + idx];
        }
}

extern "C" void kernel_launch(void* const* d_in, const int* in_sizes, int n_in,
                              void* d_out, int out_size, void* d_ws, size_t ws_size,
                              hipStream_t stream) {
    const float* h_re  = (const float*)d_in[0];
    const float* h_im  = (const float*)d_in[1];
    const float* cb_re = (const float*)d_in[2];
    const float* cb_im = (const float*)d_in[3];
    float* ws = (float*)d_ws;
    unsigned long long* packed = (unsigned long long*)(ws + OFF_PACK);

    prep_kernel<<<17, 256, 0, stream>>>(h_re, h_im, cb_re, cb_im, ws);
    pmi_wmma_kernel<<<128 * 32, 32, 0, stream>>>(ws, packed);

    int*   pmi_out  = (int*)d_out;
    float* fo       = (float*)d_out;
    finalize_kernel<<<1, 256, 0, stream>>>(ws, pmi_out, fo + NRB, fo + 2*NRB);
}